// ProbAttention_1726576856564
// MI455X (gfx1250) — compile-verified
//
#include <hip/hip_runtime.h>
#include <hip/hip_bf16.h>
#include <math.h>

// ---------------- problem constants (fixed by setup_inputs) ----------------
constexpr int B = 2;
constexpr int L = 4096;
constexpr int H = 8;
constexpr int D = 64;
constexpr int U = 45;        // FACTOR * ceil(ln(4096)) = 5*9
constexpr int UPAD = 48;     // 3 tiles of 16 rows

typedef __attribute__((ext_vector_type(2))) float v2f;
typedef __attribute__((ext_vector_type(8))) float v8f;

// D = A(16x4) * B(4x16) + C, all f32, wave32 — CDNA5 WMMA
__device__ __forceinline__ v8f wmma4(v2f a, v2f b, v8f c) {
  return __builtin_amdgcn_wmma_f32_16x16x4_f32(
      /*neg_a=*/false, a, /*neg_b=*/false, b,
      /*c_mod=*/(short)0, c, /*reuse_a=*/false, /*reuse_b=*/false);
}

// ---------------------------------------------------------------------------
// Stage 1: M[b,h,l] = max_s(Q[l].K[idx[l,s]]) - sum_s(...)/L
// grid: B*H*L/256, block: 256. One thread per query.
// ---------------------------------------------------------------------------
__global__ void k_sample_scores(const float* __restrict__ Q,
                                const float* __restrict__ K,
                                const int* __restrict__ idx,
                                float* __restrict__ M) {
  int gid = blockIdx.x * blockDim.x + threadIdx.x;   // == (b*H + h)*L + l
  if (gid >= B * H * L) return;
  int l = gid % L;
  int h = (gid / L) % H;
  int b = gid / (L * H);

  const float4* q4 = reinterpret_cast<const float4*>(
      Q + ((size_t)(b * L + l) * H + h) * D);
  float4 qv[16];
#pragma unroll
  for (int i = 0; i < 16; ++i) qv[i] = q4[i];

  float mx = -INFINITY, sm = 0.f;
  for (int s = 0; s < U; ++s) {
    int ks = idx[l * U + s];
    const float4* k4 = reinterpret_cast<const float4*>(
        K + ((size_t)(b * L + ks) * H + h) * D);
    float acc = 0.f;
#pragma unroll
    for (int i = 0; i < 16; ++i) {
      float4 kv = k4[i];
      acc += qv[i].x * kv.x + qv[i].y * kv.y + qv[i].z * kv.z + qv[i].w * kv.w;
    }
    mx = fmaxf(mx, acc);
    sm += acc;
  }
  M[gid] = mx - sm * (1.0f / (float)L);
}

// ---------------------------------------------------------------------------
// Stage 2: per (b,h) pick top-45 indices of M (ties -> lowest index).
// grid: B*H blocks, 256 threads.
// ---------------------------------------------------------------------------
__global__ void k_topk(const float* __restrict__ M, int* __restrict__ Mtop) {
  __shared__ float sM[L];
  __shared__ float sval[256];
  __shared__ int sidx[256];
  int bh = blockIdx.x;
  int tid = threadIdx.x;

  for (int i = tid; i < L; i += 256) sM[i] = M[(size_t)bh * L + i];
  __syncthreads();

  for (int t = 0; t < U; ++t) {
    float best = -INFINITY;
    int bidx = L;
    for (int i = tid; i < L; i += 256) {
      float v = sM[i];
      if (v > best) { best = v; bidx = i; }
    }
    sval[tid] = best;
    sidx[tid] = bidx;
    __syncthreads();
    for (int off = 128; off > 0; off >>= 1) {
      if (tid < off) {
        float ov = sval[tid + off]; int oi = sidx[tid + off];
        if (ov > sval[tid] || (ov == sval[tid] && oi < sidx[tid])) {
          sval[tid] = ov; sidx[tid] = oi;
        }
      }
      __syncthreads();
    }
    if (tid == 0) {
      Mtop[bh * U + t] = sidx[0];
      sM[sidx[0]] = -INFINITY;
    }
    __syncthreads();
  }
}

// ---------------------------------------------------------------------------
// Stage 3: out[b,h,:,:] = cumsum(V, axis=L). grid: B*H blocks, 256 threads:
// 64 columns x 4 segments of 1024 rows; segmented scan (read twice, write once)
// ---------------------------------------------------------------------------
__global__ void k_cumsum(const float* __restrict__ V, float* __restrict__ out) {
  __shared__ float tot[4][64];
  int bh = blockIdx.x;
  int b = bh >> 3, h = bh & 7;
  int tid = threadIdx.x;
  int seg = tid >> 6;
  int d = tid & 63;
  constexpr int SR = L / 4;
  int r0 = seg * SR;

  size_t vbase = ((size_t)b * L * H + h) * D + d;   // + r*H*D
  float t = 0.f;
  for (int r = 0; r < SR; ++r) t += V[vbase + (size_t)(r0 + r) * (H * D)];
  tot[seg][d] = t;
  __syncthreads();

  float acc = 0.f;
  for (int s = 0; s < seg; ++s) acc += tot[s][d];

  size_t obase = (size_t)bh * L * D + d;            // + r*D
  for (int r = 0; r < SR; ++r) {
    acc += V[vbase + (size_t)(r0 + r) * (H * D)];
    out[obase + (size_t)(r0 + r) * D] = acc;
  }
}

// ---------------------------------------------------------------------------
// Stage 4: flash-style masked attention for the 45 selected queries,
// scattered into out. grid: B*H blocks, 96 threads (3 waves, 16 rows each).
// Both GEMMs via v_wmma_f32_16x16x4_f32.
// ---------------------------------------------------------------------------
__global__ __launch_bounds__(96) void k_attn(const float* __restrict__ Q,
                                             const float* __restrict__ Kg,
                                             const float* __restrict__ Vg,
                                             const int* __restrict__ Mtop,
                                             float* __restrict__ out) {
  __shared__ float sQ[UPAD * D];       // pre-scaled selected queries
  __shared__ float sK[16 * D];
  __shared__ float sV[16 * D];
  __shared__ float sP[3 * 16 * 16];    // per-wave prob tile (D-layout -> A-layout)
  __shared__ int sTop[UPAD];

  int bh = blockIdx.x;
  int b = bh >> 3, h = bh & 7;
  int tid = threadIdx.x;
  int wave = tid >> 5;
  int lane = tid & 31;

  if (tid < UPAD) sTop[tid] = (tid < U) ? Mtop[bh * U + tid] : 0;
  __syncthreads();

  for (int i = tid; i < UPAD * D; i += 96) {
    int r = i >> 6, d = i & 63;
    sQ[i] = (r < U)
        ? Q[((size_t)(b * L + sTop[r]) * H + h) * D + d] * 0.125f  // 1/sqrt(64)
        : 0.0f;
  }
  __syncthreads();

  const int ln = lane & 15;            // A: row m / B: col n / C,D: col n
  const int koff = (lane >> 4) << 1;   // K-pair selector for A/B frags
  const int rowbase = (lane >> 4) << 3;// C/D rows rowbase..rowbase+7

  // Q A-fragments held in VGPRs for the whole sweep
  v2f aq[16];
#pragma unroll
  for (int c = 0; c < 16; ++c) {
    aq[c].x = sQ[(wave * 16 + ln) * D + 4 * c + koff];
    aq[c].y = sQ[(wave * 16 + ln) * D + 4 * c + koff + 1];
  }

  int qi[8];
  float mrun[8], lrun[8];
  v8f o[4];
#pragma unroll
  for (int v = 0; v < 8; ++v) {
    qi[v] = sTop[wave * 16 + rowbase + v];
    mrun[v] = -INFINITY;
    lrun[v] = 0.f;
  }
#pragma unroll
  for (int c = 0; c < 4; ++c)
#pragma unroll
    for (int v = 0; v < 8; ++v) o[c][v] = 0.f;

  for (int kt = 0; kt < L / 16; ++kt) {
    __syncthreads();   // all waves done with previous sK/sV
    for (int i = tid; i < 16 * D; i += 96) {
      int r = i >> 6, d = i & 63;
      size_t g = ((size_t)(b * L + kt * 16 + r) * H + h) * D + d;
      sK[i] = Kg[g];
      sV[i] = Vg[g];
    }
    __syncthreads();

    // S = Qtile(16x64) @ Ktile^T(64x16)  -- 16 chained K=4 WMMAs
    v8f s;
#pragma unroll
    for (int v = 0; v < 8; ++v) s[v] = 0.f;
#pragma unroll
    for (int c = 0; c < 16; ++c) {
      v2f bb;
      bb.x = sK[ln * D + 4 * c + koff];
      bb.y = sK[ln * D + 4 * c + koff + 1];
      s = wmma4(aq[c], bb, s);
    }

    // causal mask + online softmax (row spans 16 lanes of a half-wave)
    int keyn = kt * 16 + ln;
#pragma unroll
    for (int v = 0; v < 8; ++v) {
      float sv = (keyn > qi[v]) ? -INFINITY : s[v];
      float x = sv;
      x = fmaxf(x, __shfl_xor(x, 1));
      x = fmaxf(x, __shfl_xor(x, 2));
      x = fmaxf(x, __shfl_xor(x, 4));
      x = fmaxf(x, __shfl_xor(x, 8));
      float nm = fmaxf(mrun[v], x);
      float corr = __expf(mrun[v] - nm);
      float p = __expf(sv - nm);
      float ps = p;
      ps += __shfl_xor(ps, 1);
      ps += __shfl_xor(ps, 2);
      ps += __shfl_xor(ps, 4);
      ps += __shfl_xor(ps, 8);
      lrun[v] = lrun[v] * corr + ps;
      mrun[v] = nm;
      o[0][v] *= corr; o[1][v] *= corr; o[2][v] *= corr; o[3][v] *= corr;
      sP[(wave * 16 + rowbase + v) * 16 + ln] = p;   // D-layout -> LDS
    }
    __syncthreads();

    // O += P(16x16) @ Vtile(16x64)  -- A frags re-read from LDS in A-layout
#pragma unroll
    for (int c2 = 0; c2 < 4; ++c2) {
#pragma unroll
      for (int kc = 0; kc < 4; ++kc) {
        v2f ap, bv;
        ap.x = sP[(wave * 16 + ln) * 16 + 4 * kc + koff];
        ap.y = sP[(wave * 16 + ln) * 16 + 4 * kc + koff + 1];
        bv.x = sV[(4 * kc + koff) * D + c2 * 16 + ln];
        bv.y = sV[(4 * kc + koff + 1) * D + c2 * 16 + ln];
        o[c2] = wmma4(ap, bv, o[c2]);
      }
    }
  }

  // scatter normalized rows (skip padding rows >= U)
#pragma unroll
  for (int v = 0; v < 8; ++v) {
    int gr = wave * 16 + rowbase + v;
    if (gr < U) {
      float inv = 1.0f / lrun[v];
      size_t obase = ((size_t)bh * L + (size_t)qi[v]) * D;
      out[obase + 0 * 16 + ln] = o[0][v] * inv;
      out[obase + 1 * 16 + ln] = o[1][v] * inv;
      out[obase + 2 * 16 + ln] = o[2][v] * inv;
      out[obase + 3 * 16 + ln] = o[3][v] * inv;
    }
  }
}

// ---------------------------------------------------------------------------
extern "C" void kernel_launch(void* const* d_in, const int* in_sizes, int n_in,
                              void* d_out, int out_size, void* d_ws, size_t ws_size,
                              hipStream_t stream) {
  const float* Q = (const float*)d_in[0];
  const float* K = (const float*)d_in[1];
  const float* V = (const float*)d_in[2];
  const int* idx = (const int*)d_in[3];
  // d_in[4] = attn_mask scalar, unused (== 0)
  float* out = (float*)d_out;

  float* Mbuf = (float*)d_ws;                                  // B*H*L floats
  int* Mtop = (int*)((char*)d_ws + (size_t)B * H * L * 4);     // B*H*U ints

  k_sample_scores<<<(B * H * L) / 256, 256, 0, stream>>>(Q, K, idx, Mbuf);
  k_topk<<<B * H, 256, 0, stream>>>(Mbuf, Mtop);
  k_cumsum<<<B * H, 256, 0, stream>>>(V, out);
  k_attn<<<B * H, 96, 0, stream>>>(Q, K, V, Mtop, out);
}